// NavierStokesLoss_14929306321125
// MI455X (gfx1250) — compile-verified
//
#include <hip/hip_runtime.h>
#include <math.h>

typedef float v2f __attribute__((ext_vector_type(2)));
typedef float v8f __attribute__((ext_vector_type(8)));

#define NPTS 65536
#define HDIM 256
#define PTS_PER_WG 16
#define VSTRIDE 258              // padded k-stride (floats): conflict-free, keeps 8B alignment
#define NVEC 8                   // h, dh_x..dh_t, d2h_x..d2h_z

// dynamic LDS partition (in floats)
#define LDSV_OFF   0
#define LDSV_SIZE  (NVEC * PTS_PER_WG * VSTRIDE)   // 33024
#define PANEL_OFF  (LDSV_OFF + LDSV_SIZE)
#define PANEL_SIZE (16 * VSTRIDE)                  // 4128
#define CHAIN_OFF  (PANEL_OFF + PANEL_SIZE)
#define CHAIN_SIZE (NVEC * PTS_PER_WG * 16)        // 2048
#define SMEM_FLOATS (CHAIN_OFF + CHAIN_SIZE)       // 39200 floats = 156800 B

extern "C" __global__ void ns_init(float* ws) {
    if (threadIdx.x == 0) { ws[0] = 0.0f; ws[1] = 0.0f; }
}

extern "C" __global__ void __launch_bounds__(256)
ns_pinn_main(const float* __restrict__ pts, const float* __restrict__ times,
             const float* __restrict__ visc,
             const float* __restrict__ W1, const float* __restrict__ b1,
             const float* __restrict__ W2, const float* __restrict__ b2,
             const float* __restrict__ W3, const float* __restrict__ b3,
             float* __restrict__ partials)
{
    extern __shared__ float smem[];
    float* ldsV  = smem + LDSV_OFF;    // [NVEC*16][VSTRIDE] layer-1 jet vectors
    float* panel = smem + PANEL_OFF;   // [16 cols][VSTRIDE] W2 column panel (col-major)
    float* chain = smem + CHAIN_OFF;   // [NVEC][16 pts][16 cols] GEMM tile exchange

    const int t  = threadIdx.x;
    const int wg = blockIdx.x;

    // ---------------- Phase A: layer-1 jets into LDS ----------------
    {
        const int p  = t >> 4;
        const int kl = t & 15;
        const int pg = wg * PTS_PER_WG + p;
        const float z0 = pts[pg * 3 + 0];
        const float z1 = pts[pg * 3 + 1];
        const float z2 = pts[pg * 3 + 2];
        const float z3 = times[pg];
        for (int kb = 0; kb < 16; ++kb) {
            const int k = kb * 16 + kl;
            const float w0 = W1[0 * HDIM + k];
            const float w1 = W1[1 * HDIM + k];
            const float w2 = W1[2 * HDIM + k];
            const float w3 = W1[3 * HDIM + k];
            const float a  = fmaf(z0, w0, fmaf(z1, w1, fmaf(z2, w2, fmaf(z3, w3, b1[k]))));
            const float h  = tanhf(a);
            const float t1 = 1.0f - h * h;
            const float m2 = -2.0f * h * t1;
            float* base = &ldsV[p * VSTRIDE + k];
            base[0 * 16 * VSTRIDE] = h;            // primal
            base[1 * 16 * VSTRIDE] = t1 * w0;      // dh/dx
            base[2 * 16 * VSTRIDE] = t1 * w1;      // dh/dy
            base[3 * 16 * VSTRIDE] = t1 * w2;      // dh/dz
            base[4 * 16 * VSTRIDE] = t1 * w3;      // dh/dt
            base[5 * 16 * VSTRIDE] = m2 * w0 * w0; // d2h/dx2
            base[6 * 16 * VSTRIDE] = m2 * w1 * w1; // d2h/dy2
            base[7 * 16 * VSTRIDE] = m2 * w2 * w2; // d2h/dz2
        }
    }
    __syncthreads();

    // per-thread layer-3 accumulators for (point = t/16, col = t%16) over all 16 tiles
    float out_acc[5]   = {0.f, 0.f, 0.f, 0.f, 0.f};
    float dout_acc[16] = {0.f};   // J[o][i] partials at [i*4+o]
    float lap_acc[3]   = {0.f, 0.f, 0.f};

    const int wv = t >> 5;                  // wave id == jet-vector index
    const int L  = t & 31;
    const int pl = L & 15;
    const int h2 = L >> 4;                  // 0: K={0,1}, 1: K={2,3}
    const unsigned aBase = (unsigned)(wv * 16 + pl) * VSTRIDE;
    const unsigned bBase = (unsigned)pl * VSTRIDE;

    for (int j = 0; j < 16; ++j) {
        // ---- stage W2[:, j*16 : j*16+16] into LDS, column-major, padded ----
        {
            const int n  = t & 15;
            const int kr = t >> 4;
            #pragma unroll
            for (int r = 0; r < 16; ++r) {
                const int k = r * 16 + kr;
                panel[n * VSTRIDE + k] = W2[k * HDIM + j * 16 + n];
            }
        }
        __syncthreads();

        // ---- wave GEMM: V_wv (16x256) @ W2 panel (256x16) via f32 WMMA ----
        // Two independent accumulator chains to break the WMMA RAW dependency.
        v8f c0 = {0.f, 0.f, 0.f, 0.f, 0.f, 0.f, 0.f, 0.f};
        v8f c1 = {0.f, 0.f, 0.f, 0.f, 0.f, 0.f, 0.f, 0.f};
        #pragma unroll 8
        for (int kk = 0; kk < 64; kk += 2) {
            const int kb0 = kk * 4 + 2 * h2;
            const int kb1 = kb0 + 4;
            v2f a0 = *reinterpret_cast<const v2f*>(&ldsV[aBase + kb0]);
            v2f b0 = *reinterpret_cast<const v2f*>(&panel[bBase + kb0]);
            v2f a1 = *reinterpret_cast<const v2f*>(&ldsV[aBase + kb1]);
            v2f b1v = *reinterpret_cast<const v2f*>(&panel[bBase + kb1]);
            c0 = __builtin_amdgcn_wmma_f32_16x16x4_f32(
                     false, a0, false, b0, (short)0, c0, false, false);
            c1 = __builtin_amdgcn_wmma_f32_16x16x4_f32(
                     false, a1, false, b1v, (short)0, c1, false, false);
        }
        // C layout: VGPR r -> (M = r + 8*h2, N = pl)
        {
            float* cb = &chain[wv * 256 + (h2 * 8) * 16 + pl];
            #pragma unroll
            for (int r = 0; r < 8; ++r) cb[r * 16] = c0[r] + c1[r];
        }
        __syncthreads();

        // ---- tanh chain + layer-3 accumulation at (point p, global col) ----
        {
            const int p    = t >> 4;
            const int ci   = t & 15;
            const int gcol = j * 16 + ci;
            float bvv[8];
            #pragma unroll
            for (int v = 0; v < 8; ++v) bvv[v] = chain[v * 256 + p * 16 + ci];
            const float bb = bvv[0] + b2[gcol];
            const float g  = tanhf(bb);
            const float t2 = 1.0f - g * g;
            float w3v[5];
            #pragma unroll
            for (int o = 0; o < 5; ++o) w3v[o] = W3[gcol * 5 + o];
            #pragma unroll
            for (int o = 0; o < 5; ++o) out_acc[o] = fmaf(g, w3v[o], out_acc[o]);
            #pragma unroll
            for (int i = 0; i < 4; ++i) {
                const float dg = t2 * bvv[1 + i];
                #pragma unroll
                for (int o = 0; o < 4; ++o)
                    dout_acc[i * 4 + o] = fmaf(dg, w3v[o], dout_acc[i * 4 + o]);
            }
            float d2sum = 0.f;
            #pragma unroll
            for (int i = 0; i < 3; ++i) {
                const float db = bvv[1 + i];
                d2sum += t2 * bvv[5 + i] - 2.0f * g * t2 * db * db;
            }
            #pragma unroll
            for (int o = 0; o < 3; ++o) lap_acc[o] = fmaf(d2sum, w3v[o], lap_acc[o]);
        }
        __syncthreads();
    }

    // ---------------- Phase D: reduce over columns, residuals, loss ----------------
    float* red  = smem;               // reuse jet region: 256 threads * 24 floats
    float* red2 = smem + PANEL_OFF;   // reuse panel region: 16 pts * 24 floats
    {
        float* my = &red[t * 24];
        #pragma unroll
        for (int o = 0; o < 5; ++o)  my[o] = out_acc[o];
        #pragma unroll
        for (int i = 0; i < 16; ++i) my[5 + i] = dout_acc[i];
        #pragma unroll
        for (int i = 0; i < 3; ++i)  my[21 + i] = lap_acc[i];
    }
    __syncthreads();
    {
        const int p = t >> 4;
        const int a = t & 15;
        float s = 0.f;
        for (int ci = 0; ci < 16; ++ci) s += red[(p * 16 + ci) * 24 + a];
        red2[p * 24 + a] = s;
        if (a < 8) {
            const int a2 = 16 + a;
            float s2 = 0.f;
            for (int ci = 0; ci < 16; ++ci) s2 += red[(p * 16 + ci) * 24 + a2];
            red2[p * 24 + a2] = s2;
        }
    }
    __syncthreads();
    if (t < PTS_PER_WG) {
        const int p  = t;
        const int pg = wg * PTS_PER_WG + p;
        float A[24];
        #pragma unroll
        for (int a = 0; a < 24; ++a) A[a] = red2[p * 24 + a];
        float outv[5];
        #pragma unroll
        for (int o = 0; o < 5; ++o) outv[o] = A[o] + b3[o];
        const float u0 = outv[0], u1 = outv[1], u2 = outv[2];
        const float rho = 1000.0f * (1.0f + 0.1f * tanhf(outv[4]));
        const float nu  = visc[pg];
        float msum = 0.f;
        #pragma unroll
        for (int cd = 0; cd < 3; ++cd) {
            const float du_dt = A[5 + 3 * 4 + cd];                 // J[cd][3]
            const float conv  = u0 * A[5 + 0 * 4 + cd]
                              + u1 * A[5 + 1 * 4 + cd]
                              + u2 * A[5 + 2 * 4 + cd];
            const float gradp = A[5 + cd * 4 + 3];                 // J[3][cd]
            const float grav  = (cd == 1) ? -9.81f : 0.0f;
            const float R = du_dt + conv + gradp / rho - nu * A[21 + cd] - grav;
            msum += R * R;
        }
        const float Rc = A[5 + 0] + A[5 + 1 * 4 + 1] + A[5 + 2 * 4 + 2];
        atomicAdd(&partials[0], msum);
        atomicAdd(&partials[1], Rc * Rc);
    }
}

extern "C" __global__ void ns_fin(const float* ws, float* out) {
    if (threadIdx.x == 0) {
        const float m = ws[0] / (float)NPTS;
        const float c = ws[1] / (float)NPTS;
        out[0] = m + 10.0f * c;
        out[1] = m;
        out[2] = c;
    }
}

extern "C" void kernel_launch(void* const* d_in, const int* in_sizes, int n_in,
                              void* d_out, int out_size, void* d_ws, size_t ws_size,
                              hipStream_t stream) {
    const float* pts   = (const float*)d_in[0];
    const float* times = (const float*)d_in[1];
    const float* visc  = (const float*)d_in[2];
    const float* W1    = (const float*)d_in[3];
    const float* b1    = (const float*)d_in[4];
    const float* W2    = (const float*)d_in[5];
    const float* b2    = (const float*)d_in[6];
    const float* W3    = (const float*)d_in[7];
    const float* b3    = (const float*)d_in[8];
    float* ws  = (float*)d_ws;
    float* out = (float*)d_out;

    const size_t smem_bytes = (size_t)SMEM_FLOATS * sizeof(float);
    hipFuncSetAttribute((const void*)ns_pinn_main,
                        hipFuncAttributeMaxDynamicSharedMemorySize,
                        (int)smem_bytes);

    ns_init<<<1, 64, 0, stream>>>(ws);
    ns_pinn_main<<<NPTS / PTS_PER_WG, 256, smem_bytes, stream>>>(
        pts, times, visc, W1, b1, W2, b2, W3, b3, ws);
    ns_fin<<<1, 64, 0, stream>>>(ws, out);
}